// NeighborNorm_10153302687998
// MI455X (gfx1250) — compile-verified
//
#include <hip/hip_runtime.h>

#define N_NODES 50000
#define N_EDGES 1250000
#define NF 64
#define EPSV 1e-5f

typedef __attribute__((ext_vector_type(2))) float v2f;
typedef __attribute__((ext_vector_type(8))) float v8f;

// -------- Pass 1: per-edge s = sum_f x[col], q = sum_f x[col]^2 via WMMA ----
// One wave handles a tile of 16 edges. A-matrix 16x4 f32 per chunk:
//   lanes 0-15  -> M=lane,    K = {4c+0, 4c+1}
//   lanes 16-31 -> M=lane-16, K = {4c+2, 4c+3}
// B = ones(4x16)  =>  D[m,n] = rowsum(m) for every n; accumulate 16 chunks in C.
__global__ __launch_bounds__(256) void edge_pass1(
    const float* __restrict__ x,
    const int* __restrict__ rowp, const int* __restrict__ colp,
    float* __restrict__ nodeS, float* __restrict__ nodeQ,
    float* __restrict__ cnt) {
  const int lane = threadIdx.x & 31;
  const int half = lane >> 4;     // which K-pair this lane supplies
  const int mrow = lane & 15;     // edge-within-tile (matrix row M)
  const int waveId = (blockIdx.x * blockDim.x + threadIdx.x) >> 5;
  const int nWaves = (gridDim.x * blockDim.x) >> 5;
  const int nTiles = N_EDGES / 16;   // 78125, exact

  v2f ones; ones.x = 1.0f; ones.y = 1.0f;

  for (int t = waveId; t < nTiles; t += nWaves) {
    const int e = t * 16 + mrow;
    const int c0 = colp[e];
    const long base = (long)c0 * NF + half * 2;

    v8f accS = {};
    v8f accQ = {};
#pragma unroll
    for (int c = 0; c < 16; ++c) {
      const float2 v = *(const float2*)(x + base + c * 4);
      v2f a;  a.x  = v.x;        a.y  = v.y;
      v2f a2; a2.x = v.x * v.x;  a2.y = v.y * v.y;
      accS = __builtin_amdgcn_wmma_f32_16x16x4_f32(
          false, a,  false, ones, (short)0, accS, false, false);
      accQ = __builtin_amdgcn_wmma_f32_16x16x4_f32(
          false, a2, false, ones, (short)0, accQ, false, false);
    }
    // D layout (N=0 column): lane 0 holds rowsum(M=0..7) in accS[0..7],
    // lane 16 holds rowsum(M=8..15). Those two lanes scatter to nodes.
    if (mrow == 0) {
      const int eb = t * 16 + half * 8;
#pragma unroll
      for (int j = 0; j < 8; ++j) {
        const int r = rowp[eb + j];
        atomicAdd(&nodeS[r], accS[j]);
        atomicAdd(&nodeQ[r], accQ[j]);
        atomicAdd(&cnt[r], 1.0f);
      }
    }
  }
}

// -------- Pass 2: per-node mean / inv-std (tiny, N=50000) -------------------
__global__ __launch_bounds__(256) void node_pass(
    const float* __restrict__ nodeS, const float* __restrict__ nodeQ,
    const float* __restrict__ cnt,
    float* __restrict__ mu, float* __restrict__ inv) {
  const int n = blockIdx.x * blockDim.x + threadIdx.x;
  if (n >= N_NODES) return;
  const float c = cnt[n];
  const float denom = fmaxf(c, 1.0f) * (float)NF;
  const float S = nodeS[n];
  const float Q = nodeQ[n];
  const float m = S / denom;
  const float var = (Q - 2.0f * m * S + (float)NF * m * m * c) / denom;
  mu[n] = m;
  inv[n] = rsqrtf(var + EPSV);
}

// -------- Pass 3: out[e,f] = gamma*( (x[col,f]-mu[row]) * inv[row] ) + beta --
// Wave-per-edge: lane covers 2 features -> 256B coalesced read + NT b64 store.
__global__ __launch_bounds__(256) void edge_pass2(
    const float* __restrict__ x,
    const int* __restrict__ rowp, const int* __restrict__ colp,
    const float* __restrict__ mu, const float* __restrict__ inv,
    const float* __restrict__ gamma, const float* __restrict__ beta,
    float* __restrict__ out) {
  const int lane = threadIdx.x & 31;
  const int waveId = (blockIdx.x * blockDim.x + threadIdx.x) >> 5;
  const int nWaves = (gridDim.x * blockDim.x) >> 5;

  const float2 g = *(const float2*)(gamma + lane * 2);
  const float2 b = *(const float2*)(beta + lane * 2);

  for (int e = waveId; e < N_EDGES; e += nWaves) {
    // prefetch next iteration's index cacheline (global_prefetch_b8)
    if (e + nWaves < N_EDGES) {
      __builtin_prefetch(rowp + e + nWaves, 0, 0);
      __builtin_prefetch(colp + e + nWaves, 0, 0);
    }
    const int r = rowp[e];
    const int c = colp[e];
    const float2 v = *(const float2*)(x + (long)c * NF + lane * 2);
    const float m = mu[r];
    const float s = inv[r];
    union { float f[2]; unsigned long long u; } cv;
    cv.f[0] = g.x * ((v.x - m) * s) + b.x;
    cv.f[1] = g.y * ((v.y - m) * s) + b.y;
    __builtin_nontemporal_store(
        cv.u, (unsigned long long*)(out + (long)e * NF + lane * 2));
  }
}

extern "C" void kernel_launch(void* const* d_in, const int* in_sizes, int n_in,
                              void* d_out, int out_size, void* d_ws, size_t ws_size,
                              hipStream_t stream) {
  const float* x     = (const float*)d_in[0];
  const int*   ei    = (const int*)d_in[1];
  const float* gamma = (const float*)d_in[2];
  const float* beta  = (const float*)d_in[3];
  const int* rowp = ei;            // edge_index[0]
  const int* colp = ei + N_EDGES;  // edge_index[1]

  float* ws    = (float*)d_ws;     // 5*N floats = 1.0 MB
  float* nodeS = ws;
  float* nodeQ = ws + 1 * N_NODES;
  float* cnt   = ws + 2 * N_NODES;
  float* mu    = ws + 3 * N_NODES;
  float* inv   = ws + 4 * N_NODES;

  hipMemsetAsync(nodeS, 0, (size_t)3 * N_NODES * sizeof(float), stream);

  edge_pass1<<<2048, 256, 0, stream>>>(x, rowp, colp, nodeS, nodeQ, cnt);
  node_pass<<<(N_NODES + 255) / 256, 256, 0, stream>>>(nodeS, nodeQ, cnt, mu, inv);
  edge_pass2<<<4096, 256, 0, stream>>>(x, rowp, colp, mu, inv, gamma, beta,
                                       (float*)d_out);
}